// AttentionConv_20796231647301
// MI455X (gfx1250) — compile-verified
//
#include <hip/hip_runtime.h>

typedef __attribute__((ext_vector_type(2))) float v2f;
typedef __attribute__((ext_vector_type(8))) float v8f;

#define BN 8
#define CI 64
#define CO 128
#define HH 128
#define WW 128
#define HWSZ (HH * WW)

// ---------------------------------------------------------------------------
// Stage 1: 1x1 conv as GEMM via V_WMMA_F32_16X16X4_F32.
//   M = 16 pixels (consecutive w), N = 16 output channels, K = 64 in 16 steps.
//   One wave -> 16 pixels x 128 channels. A fragments loaded once, reused for
//   all 8 N-tiles. Output written NHWC to workspace (coalesced per half-wave).
// ---------------------------------------------------------------------------
__global__ __launch_bounds__(256)
void conv1x1_wmma_f32(const float* __restrict__ x,    // (B, Ci, H, W)
                      const float* __restrict__ Wc,   // (C, Ci)
                      const float* __restrict__ bc,   // (C,)
                      float* __restrict__ y)          // (B, H, W, C)  NHWC
{
    const int lane = threadIdx.x & 31;
    const int wave = threadIdx.x >> 5;
    const int tile = blockIdx.x * 8 + wave;      // 8192 tiles of 16 pixels
    const int w0   = (tile & 7) * 16;            // W/16 = 8 tiles per row
    const int rest = tile >> 3;
    const int h    = rest & (HH - 1);
    const int b    = rest >> 7;

    const int lane15 = lane & 15;                // A: row M   / B,C,D: col N
    const int kHalf  = (lane >> 4) * 2;          // 0 (lanes 0-15) or 2 (16-31)

    // A fragment (f32 16x4 layout): lane<16 holds K=k0,k0+1; lane>=16 K=k0+2,k0+3
    v2f a[16];
    const float* xb = x + ((size_t)b * CI) * HWSZ + (size_t)h * WW + (w0 + lane15);
#pragma unroll
    for (int s = 0; s < 16; ++s) {
        const int kA = 4 * s + kHalf;
        a[s].x = xb[(size_t)kA * HWSZ];
        a[s].y = xb[(size_t)(kA + 1) * HWSZ];
    }

    float* yb = y + (((size_t)b * HH + h) * WW + w0) * CO;

#pragma unroll
    for (int n = 0; n < 8; ++n) {
        const int c0 = n * 16;

        // C initialized with bias: every C/D element in lane L is channel c0+(L&15)
        const float bias = bc[c0 + lane15];
        v8f acc;
#pragma unroll
        for (int r = 0; r < 8; ++r) acc[r] = bias;

        // B fragment (f32 4x16): lane L holds N=L&15, K=k0+kHalf(+1) -> contiguous pair
        const float* wb = Wc + (size_t)(c0 + lane15) * CI + kHalf;
#pragma unroll
        for (int s = 0; s < 16; ++s) {
            const v2f bf = *(const v2f*)(wb + 4 * s);
            acc = __builtin_amdgcn_wmma_f32_16x16x4_f32(
                false, a[s], false, bf, (short)0, acc, false, false);
        }

        // D layout: VGPR r, lane L -> pixel M = r + 8*(L>>4), channel N = c0+(L&15)
#pragma unroll
        for (int r = 0; r < 8; ++r) {
            const int M = r + 8 * (lane >> 4);
            yb[(size_t)M * CO + (c0 + lane15)] = acc[r];
        }
    }
}

// ---------------------------------------------------------------------------
// Stage 2: 3x3 local attention. One block = 8x8 pixel tile; 10x10 halo x 128ch
// staged in LDS (stride 132 floats to break 128-float bank aliasing) using
// GLOBAL_LOAD_ASYNC_TO_LDS_B128 (memory -> LDS direct, ASYNCcnt-tracked).
// 4 lanes per pixel, 32 channels each; dot reduced with wave32 shfl_xor.
// ---------------------------------------------------------------------------
#define TILE 8
#define HALO 10
#define LSTRIDE 132

__global__ __launch_bounds__(256)
void local_attn3x3(const float* __restrict__ y,   // (B, H, W, C) NHWC
                   float* __restrict__ out)       // (B, C, H, W) NCHW
{
    __shared__ float ld[HALO * HALO * LSTRIDE];   // 52,800 B

    const int t  = threadIdx.x;
    const int bx = blockIdx.x & 15;               // W/8 = 16
    const int by = (blockIdx.x >> 4) & 15;        // H/8 = 16
    const int b  = blockIdx.x >> 8;

    // Load 10x10 positions x 128 channels, 16B per lane per iteration.
    // In-bounds: async copy straight into LDS. OOB halo: zero-fill via DS.
    for (int idx = t; idx < HALO * HALO * (CO / 4); idx += 256) {
        const int pos = idx >> 5;                 // / (CO/4)
        const int c4  = idx & 31;
        const int gy  = by * TILE - 1 + pos / HALO;
        const int gx  = bx * TILE - 1 + pos % HALO;

        float* lds_elem = &ld[pos * LSTRIDE + c4 * 4];
        if (gy >= 0 && gy < HH && gx >= 0 && gx < WW) {
            const float* gsrc =
                y + (((size_t)b * HH + gy) * WW + gx) * CO + c4 * 4;
            // 32-bit LDS byte address (addrspace(3) pointers are 32-bit)
            const unsigned loff = (unsigned)(size_t)(
                (__attribute__((address_space(3))) float*)lds_elem);
            // memory -> LDS, 16 bytes/lane, tracked by ASYNCcnt
            asm volatile("global_load_async_to_lds_b128 %0, %1, off"
                         :
                         : "v"(loff), "v"(gsrc)
                         : "memory");
        } else {
            *(float4*)lds_elem = make_float4(0.f, 0.f, 0.f, 0.f);
        }
    }
    // Drain this wave's async copies, then rendezvous with the block.
    asm volatile("s_wait_asynccnt 0x0" ::: "memory");
    __syncthreads();

    const int p  = t >> 2;                        // pixel 0..63
    const int q  = t & 3;                         // channel quarter
    const int py = p >> 3, px = p & 7;
    const int cbase = q * 32;

    float mid[32];
    const float* ctr = &ld[((py + 1) * HALO + (px + 1)) * LSTRIDE + cbase];
#pragma unroll
    for (int j = 0; j < 32; ++j) mid[j] = ctr[j];

    float sc[9];
#pragma unroll
    for (int n = 0; n < 9; ++n) {
        const float* pp = &ld[((py + n / 3) * HALO + (px + n % 3)) * LSTRIDE + cbase];
        float d = 0.f;
#pragma unroll
        for (int j = 0; j < 32; ++j) d = fmaf(pp[j], mid[j], d);
        d += __shfl_xor(d, 1, 32);                // reduce across the 4 lanes
        d += __shfl_xor(d, 2, 32);                // of this pixel (wave32)
        sc[n] = d * 0.08838834764831845f;         // 1/sqrt(128)
    }

    float m = sc[0];
#pragma unroll
    for (int n = 1; n < 9; ++n) m = fmaxf(m, sc[n]);
    float ssum = 0.f;
#pragma unroll
    for (int n = 0; n < 9; ++n) { sc[n] = __expf(sc[n] - m); ssum += sc[n]; }
    const float inv = 1.f / ssum;

    float o[32];
#pragma unroll
    for (int j = 0; j < 32; ++j) o[j] = 0.f;
#pragma unroll
    for (int n = 0; n < 9; ++n) {
        const float aw = sc[n] * inv;
        const float* pp = &ld[((py + n / 3) * HALO + (px + n % 3)) * LSTRIDE + cbase];
#pragma unroll
        for (int j = 0; j < 32; ++j) o[j] = fmaf(aw, pp[j], o[j]);
    }

    const int gy = by * TILE + py;
    const int gx = bx * TILE + px;
    float* ob = out + (size_t)b * CO * HWSZ + (size_t)gy * WW + gx;
#pragma unroll
    for (int j = 0; j < 32; ++j) {
        ob[(size_t)(cbase + j) * HWSZ] = o[j];
    }
}

// ---------------------------------------------------------------------------
extern "C" void kernel_launch(void* const* d_in, const int* in_sizes, int n_in,
                              void* d_out, int out_size, void* d_ws, size_t ws_size,
                              hipStream_t stream)
{
    const float* x  = (const float*)d_in[0];   // (8, 64, 128, 128)
    const float* Wc = (const float*)d_in[1];   // (128, 64)
    const float* bc = (const float*)d_in[2];   // (128,)
    float* out = (float*)d_out;                // (8, 128, 128, 128)
    float* y   = (float*)d_ws;                 // NHWC intermediate: 64 MiB

    // 8192 pixel-tiles of 16, 8 waves/block
    conv1x1_wmma_f32<<<1024, 256, 0, stream>>>(x, Wc, bc, y);
    // 8 * 16 * 16 tiles of 8x8 pixels
    local_attn3x3<<<2048, 256, 0, stream>>>(y, out);
}